// JCombineAttn_90288802496811
// MI455X (gfx1250) — compile-verified
//
#include <hip/hip_runtime.h>

// Problem constants (match reference)
constexpr int DM = 128;          // model dim
constexpr int LM = 768;          // sequence length
constexpr int NLAYER = 6;
constexpr int HM = 8;            // heads
constexpr int DHM = 16;          // head dim
constexpr int LDSZ = LM * DM;    // elements of one [L,D] activation

typedef __attribute__((ext_vector_type(2))) float v2f;
typedef __attribute__((ext_vector_type(8))) float v8f;

__device__ __forceinline__ v2f ld2(const float* __restrict__ p) {
  v2f r; r.x = p[0]; r.y = p[1]; return r;
}

// ---------------------------------------------------------------------------
// DPP 16-lane row reductions (VALU only, no LDS/ds_bpermute).
// row_ror:n rotates within each 16-lane DPP row; rotations by 1,2,4,8 form a
// butterfly, leaving every lane of the row with the full reduction.
// ---------------------------------------------------------------------------
template<int N>
__device__ __forceinline__ float dpp_rotf(float x) {
  return __int_as_float(__builtin_amdgcn_update_dpp(
      0, __float_as_int(x), 0x120 + N /*row_ror:N*/, 0xf, 0xf, true));
}
__device__ __forceinline__ float redsum16(float x) {
  x += dpp_rotf<1>(x);
  x += dpp_rotf<2>(x);
  x += dpp_rotf<4>(x);
  x += dpp_rotf<8>(x);
  return x;
}
__device__ __forceinline__ float redmax16(float x) {
  x = fmaxf(x, dpp_rotf<1>(x));
  x = fmaxf(x, dpp_rotf<2>(x));
  x = fmaxf(x, dpp_rotf<4>(x));
  x = fmaxf(x, dpp_rotf<8>(x));
  return x;
}

// ---------------------------------------------------------------------------
// Generic fp32 WMMA GEMM: one wave computes one 16x16 tile of C.
//   C[m,n] = sum_k A'[m,k] * B'[k,n]   (+ optional epilogues)
//   TRANS_A : A' = A^T       (A accessed as A[k*lda + m])
//   B_IS_WT : B' = B^T       (torch Linear weight [N,K] -> B[n*ldb + k])
//   HAS_SCALEW: C *= SW[n*ldsw + m]  (elementwise, used for M = W.T ⊙ S)
//   HAS_BIAS  : C += bias[n]
//   HAS_VEC   : C += vec1[n]*vec2[n] (used for lo_b[o]*T[o])
//   HAS_RES   : C += Res[m*ldc + n]  (residual)
// K must be a multiple of 8 (128 / 768 here).
// Grid: (M/16, N/64), block (32,4) -> 4 waves per block, one N-tile each.
// ---------------------------------------------------------------------------
template<int TRANS_A, int B_IS_WT, int HAS_BIAS, int HAS_RES, int HAS_VEC, int HAS_SCALEW>
__global__ __launch_bounds__(128)
void wmma_gemm_kernel(const float* __restrict__ A, const float* __restrict__ Bm,
                      const float* __restrict__ bias, const float* __restrict__ Res,
                      const float* __restrict__ vec1, const float* __restrict__ vec2,
                      const float* __restrict__ SW, float* __restrict__ C,
                      int K, int lda, int ldb, int ldc, int ldsw)
{
  const int lane = threadIdx.x;
  const int col  = lane & 15;        // N (or M for A-frag) index within tile
  const int hf   = lane >> 4;        // which half of the wave
  const int kb   = hf * 2;           // K sub-offset of this half's fragment
  const int m0   = blockIdx.x * 16;
  const int n0   = (blockIdx.y * blockDim.y + threadIdx.y) * 16;

  v8f acc = {};
#pragma unroll 2
  for (int k0 = 0; k0 < K; k0 += 8) {
    // batch all 4 fragment loads, then the 2 WMMAs -> loads overlap
    v2f a0, a1, b0, b1;
    if (TRANS_A) {
      a0.x = A[(k0 + kb) * lda + m0 + col];
      a0.y = A[(k0 + kb + 1) * lda + m0 + col];
      a1.x = A[(k0 + 4 + kb) * lda + m0 + col];
      a1.y = A[(k0 + 4 + kb + 1) * lda + m0 + col];
    } else {
      a0 = ld2(A + (m0 + col) * lda + k0 + kb);
      a1 = ld2(A + (m0 + col) * lda + k0 + 4 + kb);
    }
    if (B_IS_WT) {
      b0 = ld2(Bm + (n0 + col) * ldb + k0 + kb);
      b1 = ld2(Bm + (n0 + col) * ldb + k0 + 4 + kb);
    } else {
      b0.x = Bm[(k0 + kb) * ldb + n0 + col];
      b0.y = Bm[(k0 + kb + 1) * ldb + n0 + col];
      b1.x = Bm[(k0 + 4 + kb) * ldb + n0 + col];
      b1.y = Bm[(k0 + 4 + kb + 1) * ldb + n0 + col];
    }
    acc = __builtin_amdgcn_wmma_f32_16x16x4_f32(false, a0, false, b0, 0, acc, false, false);
    acc = __builtin_amdgcn_wmma_f32_16x16x4_f32(false, a1, false, b1, 0, acc, false, false);
  }

#pragma unroll
  for (int r = 0; r < 8; ++r) {
    const int m = m0 + r + hf * 8;       // C layout: VGPR r holds rows r / r+8
    const int n = n0 + col;
    float v = acc[r];
    if (HAS_SCALEW) v *= SW[n * ldsw + m];
    if (HAS_BIAS)   v += bias[n];
    if (HAS_VEC)    v += vec1[n] * vec2[n];
    if (HAS_RES)    v += Res[m * ldc + n];
    C[m * ldc + n] = v;
  }
}

// ---------------------------------------------------------------------------
// Flash-style MHA for DH=16: one wave handles one (head, 16-query-row) tile.
// qkv is [L, 3*D] packed (q|k|v). Output o is [L, D].
// ---------------------------------------------------------------------------
__global__ __launch_bounds__(32)
void flash_attn_kernel(const float* __restrict__ qkv, float* __restrict__ o)
{
  const int h    = blockIdx.y;
  const int i0   = blockIdx.x * 16;
  const int lane = threadIdx.x;
  const int col  = lane & 15;
  const int hf   = lane >> 4;
  const int kb   = hf * 2;
  const int ldq  = 3 * DM;
  const float* Q  = qkv;
  const float* Kp = qkv + DM;
  const float* Vp = qkv + 2 * DM;
  const float scale = 0.25f;            // 1/sqrt(16)

  __shared__ float pl[16 * 17];         // padded P tile for layout transpose

  // Q fragments for the 4 K-steps of the 16-deep dot product
  v2f qa[4];
#pragma unroll
  for (int t = 0; t < 4; ++t)
    qa[t] = ld2(Q + (i0 + col) * ldq + h * DHM + 4 * t + kb);

  float mi[8], li[8];
  v8f acc = {};
#pragma unroll
  for (int r = 0; r < 8; ++r) { mi[r] = -1e30f; li[r] = 0.0f; }

  for (int j0 = 0; j0 < LM; j0 += 16) {
    // ---- scores tile S = Q @ K^T : batch-load 4 K-frags, then WMMA chain
    v2f bk[4];
#pragma unroll
    for (int t = 0; t < 4; ++t)
      bk[t] = ld2(Kp + (j0 + col) * ldq + h * DHM + 4 * t + kb);
    v8f s = {};
#pragma unroll
    for (int t = 0; t < 4; ++t)
      s = __builtin_amdgcn_wmma_f32_16x16x4_f32(false, qa[t], false, bk[t], 0, s, false, false);

    // ---- online softmax; row r+8*hf lives on this half's 16 lanes (DPP row)
#pragma unroll
    for (int r = 0; r < 8; ++r) {
      const float sv   = s[r] * scale;
      const float rm   = redmax16(sv);
      const float mnew = fmaxf(mi[r], rm);
      const float a_   = __expf(mi[r] - mnew);
      const float p    = __expf(sv - mnew);
      const float rs   = redsum16(p);
      li[r] = li[r] * a_ + rs;
      mi[r] = mnew;
      acc[r] *= a_;
      pl[(r + hf * 8) * 17 + col] = p;   // C-layout -> LDS row-major P
    }
    __syncthreads();

    // ---- acc += P @ V : batch LDS A-frags + global V-frags, then WMMA chain
    v2f pa[4], vb[4];
#pragma unroll
    for (int t = 0; t < 4; ++t) {
      pa[t].x = pl[col * 17 + 4 * t + kb];
      pa[t].y = pl[col * 17 + 4 * t + kb + 1];
      vb[t].x = Vp[(j0 + 4 * t + kb) * ldq + h * DHM + col];
      vb[t].y = Vp[(j0 + 4 * t + kb + 1) * ldq + h * DHM + col];
    }
#pragma unroll
    for (int t = 0; t < 4; ++t)
      acc = __builtin_amdgcn_wmma_f32_16x16x4_f32(false, pa[t], false, vb[t], 0, acc, false, false);
    __syncthreads();
  }

#pragma unroll
  for (int r = 0; r < 8; ++r)
    o[(i0 + r + hf * 8) * DM + h * DHM + col] = acc[r] / li[r];
}

// ---------------------------------------------------------------------------
// In-place LayerNorm over last dim (D=128): one block (128 threads) per row.
// DPP 16-lane partials + tiny LDS combine (8 groups).
// ---------------------------------------------------------------------------
__global__ __launch_bounds__(128)
void ln_kernel(float* __restrict__ y, const float* __restrict__ g, const float* __restrict__ b)
{
  const int row = blockIdx.x, tid = threadIdx.x;
  __shared__ float red[8];
  const float v = y[row * DM + tid];

  float s = redsum16(v);
  if ((tid & 15) == 0) red[tid >> 4] = s;
  __syncthreads();
  float tot = 0.0f;
#pragma unroll
  for (int i = 0; i < 8; ++i) tot += red[i];
  const float mean = tot * (1.0f / DM);
  const float d = v - mean;

  float s2 = redsum16(d * d);
  __syncthreads();
  if ((tid & 15) == 0) red[tid >> 4] = s2;
  __syncthreads();
  tot = 0.0f;
#pragma unroll
  for (int i = 0; i < 8; ++i) tot += red[i];
  const float var = tot * (1.0f / DM);

  y[row * DM + tid] = d * rsqrtf(var + 1e-5f) * g[tid] + b[tid];
}

// T[o] = sum_j tx[j,o]   (one block of 128 threads)
__global__ void colsum_kernel(const float* __restrict__ tx, float* __restrict__ T)
{
  const int c = threadIdx.x;
  float s = 0.0f;
#pragma unroll 8
  for (int j = 0; j < LM; ++j) s += tx[j * DM + c];
  T[c] = s;
}

// x += x1 + x2
__global__ void add3_kernel(float* __restrict__ x, const float* __restrict__ x1,
                            const float* __restrict__ x2, int n)
{
  const int i = blockIdx.x * blockDim.x + threadIdx.x;
  if (i < n) x[i] += x1[i] + x2[i];
}

// ---------------------------------------------------------------------------
extern "C" void kernel_launch(void* const* d_in, const int* in_sizes, int n_in,
                              void* d_out, int out_size, void* d_ws, size_t ws_size,
                              hipStream_t stream)
{
  const float* x         = (const float*)d_in[0];
  const float* cop_l_w   = (const float*)d_in[1];
  const float* cop_l_b   = (const float*)d_in[2];
  const float* cop_lo_w  = (const float*)d_in[3];
  const float* cop_lo_b  = (const float*)d_in[4];
  const float* cop_out_w = (const float*)d_in[5];
  const float* cop_out_b = (const float*)d_in[6];
  const float* cop_ln_g  = (const float*)d_in[7];
  const float* cop_ln_b  = (const float*)d_in[8];
  const float* qkv_w     = (const float*)d_in[9];
  const float* qkv_b     = (const float*)d_in[10];
  const float* mha_out_w = (const float*)d_in[11];
  const float* mha_out_b = (const float*)d_in[12];
  const float* mha_ln_g  = (const float*)d_in[13];
  const float* mha_ln_b  = (const float*)d_in[14];
  const float* l1_w      = (const float*)d_in[15];
  const float* l1_b      = (const float*)d_in[16];

  // workspace layout (floats): ~3.6 MB total
  float* ws   = (float*)d_ws;
  float* a    = ws;
  float* tx   = a    + LDSZ;
  float* y1   = tx   + LDSZ;
  float* qkv  = y1   + LDSZ;          // 3*LDSZ
  float* o    = qkv  + 3 * LDSZ;
  float* y2   = o    + LDSZ;
  float* xcur = y2   + LDSZ;
  float* S    = xcur + LDSZ;          // 128x128 (becomes M = W.T ⊙ S)
  float* T    = S    + DM * DM;       // 128

  hipMemcpyAsync(xcur, x, LDSZ * sizeof(float), hipMemcpyDeviceToDevice, stream);

  const dim3 blk(32, 4);
  const dim3 gLD(LM / 16, DM / 64);        // 48 x 2   (768x128 outputs)
  const dim3 gQKV(LM / 16, 3 * DM / 64);   // 48 x 6   (768x384)
  const dim3 gS(DM / 16, DM / 64);         // 8 x 2    (128x128)

  for (int i = 0; i < NLAYER; ++i) {
    const float* lw  = cop_l_w   + i * DM * DM;
    const float* lb  = cop_l_b   + i * DM;
    const float* low = cop_lo_w  + i * DM * DM;
    const float* lob = cop_lo_b  + i * DM;
    const float* ow  = cop_out_w + i * DM * DM;
    const float* ob  = cop_out_b + i * DM;
    const float* qw  = qkv_w     + i * 3 * DM * DM;
    const float* qb  = qkv_b     + i * 3 * DM;
    const float* mw  = mha_out_w + i * DM * DM;
    const float* mb  = mha_out_b + i * DM;

    // a = x @ cop_l_w^T + b
    wmma_gemm_kernel<0,1,1,0,0,0><<<gLD, blk, 0, stream>>>(
        xcur, lw, lb, nullptr, nullptr, nullptr, nullptr, a, DM, DM, DM, DM, 0);
    // tx = x @ cop_out_w^T + b
    wmma_gemm_kernel<0,1,1,0,0,0><<<gLD, blk, 0, stream>>>(
        xcur, ow, ob, nullptr, nullptr, nullptr, nullptr, tx, DM, DM, DM, DM, 0);
    // T = colsum(tx)
    colsum_kernel<<<1, DM, 0, stream>>>(tx, T);
    // M[c,o] = lo_w[o,c] * (a^T @ tx)[c,o]     (K = 768)
    wmma_gemm_kernel<1,0,0,0,0,1><<<gS, blk, 0, stream>>>(
        a, tx, nullptr, nullptr, nullptr, nullptr, low, S, LM, DM, DM, DM, DM);
    // y1 = x + a @ M + lo_b ⊙ T
    wmma_gemm_kernel<0,0,0,1,1,0><<<gLD, blk, 0, stream>>>(
        a, S, nullptr, xcur, lob, T, nullptr, y1, DM, DM, DM, DM, 0);
    // x1 = LN(y1)  (in place)
    ln_kernel<<<LM, DM, 0, stream>>>(y1, cop_ln_g + i * DM, cop_ln_b + i * DM);
    // qkv = x @ qkv_w^T + qkv_b
    wmma_gemm_kernel<0,1,1,0,0,0><<<gQKV, blk, 0, stream>>>(
        xcur, qw, qb, nullptr, nullptr, nullptr, nullptr, qkv, DM, DM, DM, 3 * DM, 0);
    // o = MHA(qkv)
    flash_attn_kernel<<<dim3(LM / 16, HM), 32, 0, stream>>>(qkv, o);
    // y2 = x + o @ mha_out^T + b
    wmma_gemm_kernel<0,1,1,1,0,0><<<gLD, blk, 0, stream>>>(
        o, mw, mb, xcur, nullptr, nullptr, nullptr, y2, DM, DM, DM, DM, 0);
    // x2 = LN(y2)  (in place)
    ln_kernel<<<LM, DM, 0, stream>>>(y2, mha_ln_g + i * DM, mha_ln_b + i * DM);
    // x = x + x1 + x2
    add3_kernel<<<(LDSZ + 255) / 256, 256, 0, stream>>>(xcur, y1, y2, LDSZ);
  }

  // out = x @ l1_w^T + l1_b
  wmma_gemm_kernel<0,1,1,0,0,0><<<gLD, blk, 0, stream>>>(
      xcur, l1_w, l1_b, nullptr, nullptr, nullptr, nullptr, (float*)d_out, DM, DM, DM, DM, 0);

  (void)in_sizes; (void)n_in; (void)out_size; (void)ws_size;
}